// MinDistTwoCircles_88716844466991
// MI455X (gfx1250) — compile-verified
//
#include <hip/hip_runtime.h>

// MinDistTwoCircles: per-item Adam optimization of two angles.
// Compute-bound (TRANS-op bound) elementwise kernel; WMMA is inapplicable
// (no matrix structure, true serial dependency chain per item).
// CDNA5 wins used: wave32, v_sin/v_cos/v_rsq/v_rcp/v_sqrt hardware TRANS ops
// (co-execute with VALU on CDNA5), global_prefetch_b8.

namespace {

constexpr float kB1     = 0.9f;
constexpr float kB2     = 0.999f;
constexpr float kEps    = 1e-8f;
constexpr float kInv2Pi = 0.15915494309189535f; // v_sin/v_cos take revolutions

struct F3 { float x, y, z; };

__device__ __forceinline__ F3 mk(float x, float y, float z) { return {x, y, z}; }
__device__ __forceinline__ F3 add3(F3 a, F3 b) { return {a.x + b.x, a.y + b.y, a.z + b.z}; }
__device__ __forceinline__ F3 sub3(F3 a, F3 b) { return {a.x - b.x, a.y - b.y, a.z - b.z}; }
__device__ __forceinline__ F3 scl3(F3 a, float s) { return {a.x * s, a.y * s, a.z * s}; }
__device__ __forceinline__ float dot3(F3 a, F3 b) { return a.x * b.x + a.y * b.y + a.z * b.z; }
__device__ __forceinline__ F3 cross3(F3 a, F3 b) {
    return {a.y * b.z - a.z * b.y, a.z * b.x - a.x * b.z, a.x * b.y - a.y * b.x};
}
// v / ||v||  via one v_rsq_f32 (TRANS) + 3 muls
__device__ __forceinline__ F3 norm3(F3 a) {
    float r = __builtin_amdgcn_rsqf(dot3(a, a));
    return scl3(a, r);
}
__device__ __forceinline__ float fsin(float t) { return __builtin_amdgcn_sinf(t * kInv2Pi); }
__device__ __forceinline__ float fcos(float t) { return __builtin_amdgcn_cosf(t * kInv2Pi); }

__device__ __forceinline__ F3 ld3(const float* __restrict__ p, int i) {
    return {p[3 * i + 0], p[3 * i + 1], p[3 * i + 2]};
}

} // namespace

__global__ void __launch_bounds__(256)
MinDistTwoCircles_88716844466991_kernel(const float* __restrict__ c1p,
                                        const float* __restrict__ r1p,
                                        const float* __restrict__ z1p,
                                        const float* __restrict__ c2p,
                                        const float* __restrict__ r2p,
                                        const float* __restrict__ z2p,
                                        const int*   __restrict__ nip,
                                        float*       __restrict__ out,
                                        int n) {
    const int i = blockIdx.x * blockDim.x + threadIdx.x;
    if (i >= n) return;

    // gfx1250 prefetch path (global_prefetch_b8): warm L2 for our operands.
    __builtin_prefetch(&c1p[3 * i], 0, 1);
    __builtin_prefetch(&z1p[3 * i], 0, 1);
    __builtin_prefetch(&c2p[3 * i], 0, 1);
    __builtin_prefetch(&z2p[3 * i], 0, 1);

    const int num_iter = nip[0];

    const F3 c1 = ld3(c1p, i);
    const F3 c2 = ld3(c2p, i);
    const float r1 = r1p[i];
    const float r2 = r2p[i];

    // _axes(z1): n = normalize; zref = (n.z < 0.01) ? ey : ez; x = n×zref / ||.||; y = n×x / ||.||
    const F3 n1 = norm3(ld3(z1p, i));
    const F3 zr1 = (n1.z < 0.01f) ? mk(0.f, 1.f, 0.f) : mk(0.f, 0.f, 1.f);
    const F3 x1 = norm3(cross3(n1, zr1));
    const F3 y1 = norm3(cross3(n1, x1));

    const F3 n2 = norm3(ld3(z2p, i));
    const F3 zr2 = (n2.z < 0.01f) ? mk(0.f, 1.f, 0.f) : mk(0.f, 0.f, 1.f);
    const F3 x2 = norm3(cross3(n2, zr2));
    const F3 y2 = norm3(cross3(n2, x2));

    // _init_guess
    const F3 v = norm3(sub3(c2, c1));
    const F3 v1 = norm3(sub3(v, scl3(n1, dot3(v, n1))));
    float t1 = atan2f(dot3(y1, v1), dot3(x1, v1));
    const F3 nv = mk(-v.x, -v.y, -v.z);
    const F3 v2 = norm3(sub3(nv, scl3(n2, dot3(nv, n2))));
    float t2 = atan2f(dot3(y2, v2), dot3(x2, v2));

    float bd = 99999.0f;
    int   bi = 0;
    F3 p1 = mk(0.f, 0.f, 0.f), p2 = mk(0.f, 0.f, 0.f);

    float lr = 0.1f;
    #pragma unroll 1
    for (int phase = 0; phase < 2; ++phase) {
        lr *= 0.1f;
        // m, v reset each phase; bias-correction step count restarts each phase.
        float m1 = 0.f, va1 = 0.f, m2 = 0.f, va2 = 0.f;
        float pb1 = 1.f, pb2 = 1.f; // B1^st, B2^st accumulated incrementally

        #pragma unroll 2
        for (int it = 0; it < num_iter; ++it) {
            const float co1 = fcos(t1), si1 = fsin(t1);
            const float co2 = fcos(t2), si2 = fsin(t2);

            p1 = add3(c1, scl3(add3(scl3(x1, co1), scl3(y1, si1)), r1));
            p2 = add3(c2, scl3(add3(scl3(x2, co2), scl3(y2, si2)), r2));

            const F3 diff = sub3(p1, p2);
            const float d2   = dot3(diff, diff);
            const float invd = __builtin_amdgcn_rsqf(d2); // one TRANS gives 1/d ...
            const float d    = d2 * invd;                 // ... and d

            const bool better = d < bd;
            bd = better ? d : bd;
            bi = better ? (phase * num_iter + it) : bi;

            const F3 dp1 = scl3(sub3(scl3(y1, co1), scl3(x1, si1)), r1);
            const F3 dp2 = scl3(sub3(scl3(y2, co2), scl3(x2, si2)), r2);
            const float g1 =  dot3(diff, dp1) * invd;
            const float g2 = -dot3(diff, dp2) * invd;

            m1  = kB1 * m1  + (1.f - kB1) * g1;
            va1 = kB2 * va1 + (1.f - kB2) * g1 * g1;
            m2  = kB1 * m2  + (1.f - kB1) * g2;
            va2 = kB2 * va2 + (1.f - kB2) * g2 * g2;

            pb1 *= kB1;
            pb2 *= kB2;
            const float rbc1 = __builtin_amdgcn_rcpf(1.f - pb1);
            const float rbc2 = __builtin_amdgcn_rcpf(1.f - pb2);

            const float den1 = __builtin_amdgcn_sqrtf(va1 * rbc2) + kEps;
            const float den2 = __builtin_amdgcn_sqrtf(va2 * rbc2) + kEps;
            t1 -= lr * m1 * rbc1 * __builtin_amdgcn_rcpf(den1);
            t2 -= lr * m2 * rbc1 * __builtin_amdgcn_rcpf(den2);
        }
    }

    // Outputs flat-concatenated in return order: bd, bi(int32), t1, t2, p1(N,3), p2(N,3)
    out[i] = bd;
    reinterpret_cast<int*>(out)[n + i] = bi;
    out[2 * n + i] = t1;
    out[3 * n + i] = t2;
    float* __restrict__ p1o = out + 4 * n;
    p1o[3 * i + 0] = p1.x; p1o[3 * i + 1] = p1.y; p1o[3 * i + 2] = p1.z;
    float* __restrict__ p2o = out + 7 * n;
    p2o[3 * i + 0] = p2.x; p2o[3 * i + 1] = p2.y; p2o[3 * i + 2] = p2.z;
}

extern "C" void kernel_launch(void* const* d_in, const int* in_sizes, int n_in,
                              void* d_out, int out_size, void* d_ws, size_t ws_size,
                              hipStream_t stream) {
    (void)n_in; (void)out_size; (void)d_ws; (void)ws_size;
    const float* c1 = (const float*)d_in[0];
    const float* r1 = (const float*)d_in[1];
    const float* z1 = (const float*)d_in[2];
    const float* c2 = (const float*)d_in[3];
    const float* r2 = (const float*)d_in[4];
    const float* z2 = (const float*)d_in[5];
    const int*   ni = (const int*)d_in[6];

    const int n = in_sizes[1]; // r1 has N elements
    const int block = 256;     // 8 wave32s per block
    const int grid = (n + block - 1) / block;

    MinDistTwoCircles_88716844466991_kernel<<<grid, block, 0, stream>>>(
        c1, r1, z1, c2, r2, z2, ni, (float*)d_out, n);
}